// FASTLoss_55207509622846
// MI455X (gfx1250) — compile-verified
//
#include <hip/hip_runtime.h>
#include <hip/hip_bf16.h>

typedef __attribute__((ext_vector_type(2))) float v2f;
typedef __attribute__((ext_vector_type(4))) float v4f;
typedef __attribute__((ext_vector_type(8))) float v8f;

#define NPIX (640*640)
#define NB   16
#define NKC  5
#define EPSF 1e-6f
#define HBINS 1024

// workspace layout in dwords
#define WS_KSUMS 0              // 240 f32: [b][c][{inter,ii,tt}]
#define WS_TSUMS 240            // 48  f32: [b][{inter,ii,tt}]
#define WS_NPOS  288            // 16  u32
#define WS_NNEG  304            // 16  u32 (unused; nneg derived from hist)
#define WS_THR   320            // 16  f32
#define WS_HIST  336            // 16*1024 u32
#define WS_TOTAL (336 + NB*HBINS)

// Fast sigmoid: v_exp_f32 + v_rcp_f32 (avoid IEEE div expansion).
__device__ __forceinline__ float sigmoidf(float x) {
    return __builtin_amdgcn_rcpf(1.0f + __expf(-x));
}

__global__ void zero_ws_kernel(unsigned* __restrict__ ws, int n) {
    for (int i = blockIdx.x * blockDim.x + threadIdx.x; i < n;
         i += gridDim.x * blockDim.x)
        ws[i] = 0u;
}

// ---------------------------------------------------------------------------
// Kernel-channel dice sums via f32 WMMA Gram accumulation.
// Streams: m=2c -> i_c = sigmoid(pred[b][1+c])*mask, m=2c+1 -> t_c = gtk[b][c]*mask.
// For the 16x16x4-f32 shape the register layouts of A (16x4) and B=A^T (4x16)
// coincide lane-for-lane, so D += A*A accumulates the Gram matrix with no
// shuffles. Since B==A, the K->spatial mapping is arbitrary: each lane loads a
// float4 at (base + 4*half) and the two WMMAs of the pair use (f0,f1) and
// (f2,f3), tiling 8 consecutive pixels per wave iteration with b128 loads.
// D[2c][2c+1]=sum(i*t), D[2c][2c]=sum(i*i), D[2c+1][2c+1]=sum(t*t).
// ---------------------------------------------------------------------------
__global__ void kernels_gram_wmma(const float* __restrict__ pred,
                                  const float* __restrict__ gtk,
                                  const float* __restrict__ tmask,
                                  float* __restrict__ ksums) {
    __shared__ float lred[15];
    const int b    = blockIdx.y;
    const int lane = threadIdx.x & 31;
    const int wave = threadIdx.x >> 5;
    const int wavesPerBlk = blockDim.x >> 5;
    const int m = lane & 15;   // stream (matrix row)
    const int h = lane >> 4;   // half

    if (threadIdx.x < 15) lred[threadIdx.x] = 0.0f;
    __syncthreads();

    const float* mbase = tmask + (size_t)b * NPIX;
    const float* sbase = nullptr;
    const bool active = (m < 10);
    const bool dosig  = active && ((m & 1) == 0);
    if (active) {
        int c = m >> 1;
        sbase = (m & 1) ? (gtk  + ((size_t)b * NKC + c) * NPIX)
                        : (pred + ((size_t)b * 6 + 1 + c) * NPIX);
    }

    v8f acc = {0.f, 0.f, 0.f, 0.f, 0.f, 0.f, 0.f, 0.f};
    const int wavesTotal = gridDim.x * wavesPerBlk;
    const int waveId     = blockIdx.x * wavesPerBlk + wave;
    const int step       = wavesTotal * 8;

    for (int base = waveId * 8; base < NPIX; base += step) {
        const int x0 = base + 4 * h;           // 16B-aligned
        __builtin_prefetch(mbase + x0 + step, 0, 0);
        v4f mv = *(const v4f*)(mbase + x0);
        v4f a  = {0.f, 0.f, 0.f, 0.f};
        if (active) {
            __builtin_prefetch(sbase + x0 + step, 0, 0);
            // stream data is read exactly once -> non-temporal
            v4f s = __builtin_nontemporal_load((const v4f*)(sbase + x0));
            if (dosig) {
                s.x = sigmoidf(s.x); s.y = sigmoidf(s.y);
                s.z = sigmoidf(s.z); s.w = sigmoidf(s.w);
            }
            a = s * mv;
        }
        v2f A0; A0.x = a.x; A0.y = a.y;
        v2f A1; A1.x = a.z; A1.y = a.w;
        acc = __builtin_amdgcn_wmma_f32_16x16x4_f32(
            false, A0, false, A0, (short)0, acc, false, false);
        acc = __builtin_amdgcn_wmma_f32_16x16x4_f32(
            false, A1, false, A1, (short)0, acc, false, false);
    }

    // Extract D[M][N]: M<8 -> VGPR M, lane N ; M>=8 -> VGPR M-8, lane N+16.
    // Order: j = c*3 + {0:inter(2c,2c+1), 1:ii(2c,2c), 2:tt(2c+1,2c+1)}
    constexpr int vg[15] = {0,0,1, 2,2,3, 4,4,5, 6,6,7, 0,0,1};
    constexpr int ln[15] = {1,0,1, 3,2,3, 5,4,5, 7,6,7, 25,24,25};
    float mine = 0.0f;
#pragma unroll
    for (int j = 0; j < 15; ++j) {
        float t = __shfl(acc[vg[j]], ln[j], 32);
        if (lane == j) mine = t;
    }
    if (lane < 15) atomicAdd(&lred[lane], mine);
    __syncthreads();
    if (threadIdx.x < 15)
        atomicAdd(&ksums[b * 15 + threadIdx.x], lred[threadIdx.x]);
}

// ---------------------------------------------------------------------------
// OHEM pass 1: per-image pos count + 1024-bin histogram of sigmoid scores of
// negatives (replaces argsort top-k with O(N) histogram select).
// n_neg is recovered later as the histogram total (no per-pixel counter).
// ---------------------------------------------------------------------------
__global__ void ohem_hist_kernel(const float* __restrict__ pred,
                                 const float* __restrict__ gtt,
                                 const float* __restrict__ tmask,
                                 unsigned* __restrict__ hist,
                                 unsigned* __restrict__ npos) {
    __shared__ unsigned lh[HBINS];
    __shared__ unsigned lpos;
    const int b = blockIdx.y;
    for (int i = threadIdx.x; i < HBINS; i += blockDim.x) lh[i] = 0u;
    if (threadIdx.x == 0) lpos = 0u;
    __syncthreads();

    const float* pb = pred + (size_t)b * 6 * NPIX;   // channel 0
    const float* gb = gtt  + (size_t)b * NPIX;
    const float* mb = tmask + (size_t)b * NPIX;
    const int lane = threadIdx.x & 31;
    const int stride = gridDim.x * blockDim.x;
    for (int x = blockIdx.x * blockDim.x + threadIdx.x; x < NPIX; x += stride) {
        __builtin_prefetch(pb + x + stride, 0, 0);
        float p = sigmoidf(pb[x]);
        float g = gb[x];
        float mm = mb[x];
        bool pos = (mm > 0.5f) && (g > 0.5f);
        bool neg = (mm > 0.5f) && (g <= 0.5f);
        // wave-aggregated positive count: 1 atomic per wave
        unsigned long long bb = __ballot(pos);
        if (lane == 0 && bb) atomicAdd(&lpos, (unsigned)__popcll(bb));
        if (neg) {
            int bin = (int)(p * (float)HBINS);
            bin = bin > (HBINS - 1) ? (HBINS - 1) : (bin < 0 ? 0 : bin);
            atomicAdd(&lh[bin], 1u);
        }
    }
    __syncthreads();
    for (int i = threadIdx.x; i < HBINS; i += blockDim.x)
        if (lh[i]) atomicAdd(&hist[b * HBINS + i], lh[i]);
    if (threadIdx.x == 0 && lpos) atomicAdd(&npos[b], lpos);
}

// OHEM pass 2: n_neg = histogram total; top-down scan for the score threshold
// so that #selected negatives ~= min(3*n_pos, n_neg).
__global__ void ohem_thresh_kernel(const unsigned* __restrict__ hist,
                                   const unsigned* __restrict__ npos,
                                   float* __restrict__ thr) {
    const int b = threadIdx.x;
    if (b >= NB) return;
    unsigned have = 0u;
    for (int bin = 0; bin < HBINS; ++bin) have += hist[b * HBINS + bin];
    unsigned want = 3u * npos[b];
    unsigned need = want < have ? want : have;
    float t;
    if (need == 0u) {
        t = 2.0f;  // select no negatives
    } else {
        t = 0.0f;
        unsigned cum = 0u;
        for (int bin = HBINS - 1; bin >= 0; --bin) {
            cum += hist[b * HBINS + bin];
            if (cum >= need) { t = (float)bin / (float)HBINS; break; }
        }
    }
    thr[b] = t;
}

// Text dice sums with OHEM mask: o = pos | (neg & p >= thr[b]).
__global__ void text_dice_kernel(const float* __restrict__ pred,
                                 const float* __restrict__ gtt,
                                 const float* __restrict__ tmask,
                                 const float* __restrict__ thr,
                                 float* __restrict__ tsums) {
    __shared__ float red[3];
    const int b = blockIdx.y;
    if (threadIdx.x < 3) red[threadIdx.x] = 0.0f;
    __syncthreads();

    const float th = thr[b];
    const float* pb = pred + (size_t)b * 6 * NPIX;
    const float* gb = gtt  + (size_t)b * NPIX;
    const float* mb = tmask + (size_t)b * NPIX;
    float s_it = 0.0f, s_ii = 0.0f, s_tt = 0.0f;
    const int stride = gridDim.x * blockDim.x;
    for (int x = blockIdx.x * blockDim.x + threadIdx.x; x < NPIX; x += stride) {
        __builtin_prefetch(pb + x + stride, 0, 0);
        float p = sigmoidf(pb[x]);
        float g = gb[x];
        float mm = mb[x];
        bool pos = (g > 0.5f) && (mm > 0.5f);
        bool neg = (g <= 0.5f) && (mm > 0.5f);
        bool sel = pos || (neg && (p >= th));
        if (sel) {
            s_it += p * g;
            s_ii += p * p;
            s_tt += g * g;
        }
    }
    // wave butterfly reduction, then one LDS atomic per wave per sum
#pragma unroll
    for (int off = 16; off > 0; off >>= 1) {
        s_it += __shfl_xor(s_it, off, 32);
        s_ii += __shfl_xor(s_ii, off, 32);
        s_tt += __shfl_xor(s_tt, off, 32);
    }
    if ((threadIdx.x & 31) == 0) {
        atomicAdd(&red[0], s_it);
        atomicAdd(&red[1], s_ii);
        atomicAdd(&red[2], s_tt);
    }
    __syncthreads();
    if (threadIdx.x < 3)
        atomicAdd(&tsums[b * 3 + threadIdx.x], red[threadIdx.x]);
}

// Finalize: dice terms -> means -> (loss, loss_text, loss_kernels).
__global__ void finalize_kernel(const float* __restrict__ ksums,
                                const float* __restrict__ tsums,
                                float* __restrict__ out) {
    __shared__ float sk, st;
    if (threadIdx.x == 0) { sk = 0.0f; st = 0.0f; }
    __syncthreads();
    if (threadIdx.x < NB * NKC) {
        const float* s = ksums + threadIdx.x * 3;
        float d = 1.0f - 2.0f * s[0] / (s[1] + s[2] + EPSF);
        atomicAdd(&sk, d);
    }
    if (threadIdx.x < NB) {
        const float* s = tsums + threadIdx.x * 3;
        float d = 1.0f - 2.0f * s[0] / (s[1] + s[2] + EPSF);
        atomicAdd(&st, d);
    }
    __syncthreads();
    if (threadIdx.x == 0) {
        float lk = sk / (float)(NB * NKC);
        float lt = st / (float)NB;
        out[0] = lk + 0.5f * lt;  // loss
        out[1] = lt;              // loss_text
        out[2] = lk;              // loss_kernels
    }
}

extern "C" void kernel_launch(void* const* d_in, const int* in_sizes, int n_in,
                              void* d_out, int out_size, void* d_ws, size_t ws_size,
                              hipStream_t stream) {
    const float* pred = (const float*)d_in[0];  // (16,6,640,640)
    const float* gtt  = (const float*)d_in[1];  // (16,1,640,640)
    const float* gtk  = (const float*)d_in[2];  // (16,5,640,640)
    const float* msk  = (const float*)d_in[3];  // (16,1,640,640)

    unsigned* ws    = (unsigned*)d_ws;
    float*    ksums = (float*)(ws + WS_KSUMS);
    float*    tsums = (float*)(ws + WS_TSUMS);
    unsigned* npos  = ws + WS_NPOS;
    float*    thr   = (float*)(ws + WS_THR);
    unsigned* hist  = ws + WS_HIST;

    zero_ws_kernel<<<32, 256, 0, stream>>>(ws, WS_TOTAL);
    kernels_gram_wmma<<<dim3(64, NB), 256, 0, stream>>>(pred, gtk, msk, ksums);
    ohem_hist_kernel<<<dim3(64, NB), 256, 0, stream>>>(pred, gtt, msk, hist, npos);
    ohem_thresh_kernel<<<1, 32, 0, stream>>>(hist, npos, thr);
    text_dice_kernel<<<dim3(64, NB), 256, 0, stream>>>(pred, gtt, msk, thr, tsums);
    finalize_kernel<<<1, 128, 0, stream>>>(ksums, tsums, (float*)d_out);
}